// SpnBlock_11467562680451
// MI455X (gfx1250) — compile-verified
//
#include <hip/hip_runtime.h>
#include <stdint.h>

// Problem constants (match reference: B=8, C=32, H=256, W=256, fp32)
#define BN 8
#define CN 32
#define HH 256
#define WW 256
#define NPLANE (BN * CN)          // 256 workgroups
#define TW 16                     // columns per tile
#define NTILES (WW / TW)          // 16
#define ROWSTRIDE 17              // floats per LDS row (16 data + 1 pad, TDM pad_amount=1 dword)
#define ARR_BYTES (HH * ROWSTRIDE * 4)   // 17408 B per array per buffer
#define BUF_BYTES (4 * ARR_BYTES)        // 69632 B per double-buffer slot
#define LDS_BYTES (2 * BUF_BYTES + 2 * HH * 4)  // 141312 B total

typedef uint32_t v4u __attribute__((ext_vector_type(4)));
typedef uint32_t v8u __attribute__((ext_vector_type(8)));

// Issue one TDM 2D tile load: 256 rows x 16 cols of fp32, global row stride = W,
// into LDS at lds_addr with 1-dword padding after every 16-dword row.
// D# group0 (4 SGPRs) + group1 (8 SGPRs); groups 2/3 omitted (2D tensor).
__device__ __forceinline__ void tdm_load_tile(uint64_t gaddr, uint32_t lds_addr) {
  v4u g0;
  g0[0] = 1u;                                   // count=1, is_restore=0, gather=0
  g0[1] = lds_addr;                             // LDS byte address (wave-relative)
  g0[2] = (uint32_t)gaddr;                      // global_addr[31:0]
  g0[3] = ((uint32_t)(gaddr >> 32) & 0x01FFFFFFu) | (2u << 30);  // addr[56:32] | type=2
  v8u g1;
  g1[0] = (2u << 16)        // data_size = 4 B
        | (1u << 20)        // pad_enable
        | (3u << 22);       // pad_interval: 16 dwords (one tile row); pad_amount=0 -> 1 dword
  g1[1] = ((uint32_t)TW & 0xFFFFu) << 16;       // tensor_dim0[15:0] = 16
  g1[2] = ((uint32_t)HH & 0xFFFFu) << 16;       // tensor_dim0 hi=0 | tensor_dim1[15:0] = 256
  g1[3] = ((uint32_t)TW) << 16;                 // tensor_dim1 hi=0 | tile_dim0 = 16
  g1[4] = (uint32_t)HH;                         // tile_dim1 = 256 rows, tile_dim2 = 0
  g1[5] = (uint32_t)WW;                         // tensor_dim0_stride = 256 (lo 32)
  g1[6] = 0u;                                   // stride0 hi | stride1 lo
  g1[7] = 0u;
  asm volatile("tensor_load_to_lds %0, %1" :: "s"(g0), "s"(g1) : "memory");
}

__global__ void __launch_bounds__(256)
spn_scan_kernel(const float* __restrict__ X, const float* __restrict__ G1,
                const float* __restrict__ G2, const float* __restrict__ G3,
                float* __restrict__ Out) {
  extern __shared__ float smem[];
  const int t = (int)threadIdx.x;               // row index 0..255
  const int p = (int)blockIdx.x;                // plane index 0..255

  const uint64_t planeByte = (uint64_t)p * (HH * WW) * 4u;
  const uint64_t gx = (uint64_t)X  + planeByte;
  const uint64_t ga = (uint64_t)G1 + planeByte;
  const uint64_t gb = (uint64_t)G2 + planeByte;
  const uint64_t gc = (uint64_t)G3 + planeByte;

  // Flat LDS pointer low 32 bits == wave-relative LDS byte offset (ISA §10.2 aperture).
  const uint32_t ldsBase = (uint32_t)(uintptr_t)smem;

  // h-column ping-pong buffers (after the two tile buffers)
  float* hA = smem + (2 * BUF_BYTES) / 4;
  float* hB = hA + HH;
  hA[t] = 0.0f;                                 // finite values for the clamped j=0 reads
  hB[t] = 0.0f;

  // Only wave 0 drives the TDM (TDM ignores EXEC; use a scalar-uniform branch).
  const bool leader = (__builtin_amdgcn_readfirstlane((int)threadIdx.x) == 0);
  if (leader) {                                 // prefetch tile 0 into buffer 0
    tdm_load_tile(gx, ldsBase + 0 * ARR_BYTES);
    tdm_load_tile(ga, ldsBase + 1 * ARR_BYTES);
    tdm_load_tile(gb, ldsBase + 2 * ARR_BYTES);
    tdm_load_tile(gc, ldsBase + 3 * ARR_BYTES);
  }

  const float fu = (t > 0) ? 1.0f : 0.0f;           // up-neighbor exists
  const float fd = (t < HH - 1) ? 1.0f : 0.0f;      // down-neighbor exists
  const int   iu = (t > 0) ? t - 1 : 0;             // clamped (value masked by fu)
  const int   id = (t < HH - 1) ? t + 1 : HH - 1;   // clamped (value masked by fd)

  float  hm = 0.0f;                              // h[t][j-1], carried in a register
  float* hprev = hA;                             // column j-1 (zeroed for j=0)
  float* hcur  = hB;

  for (int k = 0; k < NTILES; ++k) {
    __syncthreads();  // (A) everyone done reading buf[(k+1)&1] from tile k-1
    if (leader) {
      if (k + 1 < NTILES) {                      // prefetch tile k+1
        const uint32_t bufOff  = (uint32_t)(((k + 1) & 1) * BUF_BYTES);
        const uint64_t colByte = (uint64_t)((k + 1) * TW) * 4u;
        tdm_load_tile(gx + colByte, ldsBase + bufOff + 0 * ARR_BYTES);
        tdm_load_tile(ga + colByte, ldsBase + bufOff + 1 * ARR_BYTES);
        tdm_load_tile(gb + colByte, ldsBase + bufOff + 2 * ARR_BYTES);
        tdm_load_tile(gc + colByte, ldsBase + bufOff + 3 * ARR_BYTES);
        __builtin_amdgcn_s_wait_tensorcnt(4);    // first 4 (tile k) complete, in order
      } else {
        __builtin_amdgcn_s_wait_tensorcnt(0);
      }
    }
    __syncthreads();  // (B) tile k resident in LDS for all waves

    const int bb = (k & 1) * (BUF_BYTES / 4);
    const float* lx = smem + bb + 0 * (ARR_BYTES / 4);
    const float* la = smem + bb + 1 * (ARR_BYTES / 4);
    const float* lb = smem + bb + 2 * (ARR_BYTES / 4);
    const float* lc = smem + bb + 3 * (ARR_BYTES / 4);
    const int r = t * ROWSTRIDE;

    float outv[TW];
#pragma unroll
    for (int jj = 0; jj < TW; ++jj) {
      float xv = lx[r + jj];
      float av = la[r + jj];
      float bv = lb[r + jj];
      float cv = lc[r + jj];

      // Gate normalization: only where |g1|+|g2|+|g3| >= 1 (sum==0 case keeps gates).
      float s = fabsf(av) + fabsf(bv) + fabsf(cv);
      if (s >= 1.0f) { float rs = 1.0f / s; av *= rs; bv *= rs; cv *= rs; }

      // Zero gates whose neighbor is out of range (j==0, i==0, i==H-1).
      const float fj = (k == 0 && jj == 0) ? 0.0f : 1.0f;
      av *= fj * fu;
      bv *= fj;
      cv *= fj * fd;

      const float hu = hprev[iu];
      const float hd = hprev[id];
      const float h  = (1.0f - av - bv - cv) * xv + av * hu + bv * hm + cv * hd;

      hcur[t]  = h;
      hm       = h;
      outv[jj] = h;
      { float* tmp = hprev; hprev = hcur; hcur = tmp; }
      if (jj < TW - 1) __syncthreads();          // column-to-column dependency
    }

    // Write this tile's 16 columns of row t: contiguous 64 B -> 4x b128 stores.
    float4* dst = (float4*)(Out + (size_t)p * (HH * WW) + (size_t)t * WW + k * TW);
#pragma unroll
    for (int q = 0; q < TW / 4; ++q) {
      float4 v;
      v.x = outv[4 * q + 0];
      v.y = outv[4 * q + 1];
      v.z = outv[4 * q + 2];
      v.w = outv[4 * q + 3];
      dst[q] = v;
    }
  }
}

extern "C" void kernel_launch(void* const* d_in, const int* in_sizes, int n_in,
                              void* d_out, int out_size, void* d_ws, size_t ws_size,
                              hipStream_t stream) {
  (void)in_sizes; (void)n_in; (void)d_ws; (void)ws_size; (void)out_size;
  const float* X  = (const float*)d_in[0];
  const float* G1 = (const float*)d_in[1];
  const float* G2 = (const float*)d_in[2];
  const float* G3 = (const float*)d_in[3];
  float* Out = (float*)d_out;
  spn_scan_kernel<<<dim3(NPLANE), dim3(HH), LDS_BYTES, stream>>>(X, G1, G2, G3, Out);
}